// GINEBlock_60601988547138
// MI455X (gfx1250) — compile-verified
//
#include <hip/hip_runtime.h>

// ---------------------------------------------------------------------------
// GINE block for MI455X (gfx1250, wave32).
// - fp32 matrix math on V_WMMA_F32_16X16X4_F32
// - LDS layouts swizzled/padded for conflict-free ds_load_b64 fragments
// - edge-tile staging via GLOBAL_LOAD_ASYNC_TO_LDS_B128 (ASYNCcnt) when available
// ---------------------------------------------------------------------------

#define HID    128
#define EDIM   64
#define BN_EPS 1e-5f

typedef float v2f __attribute__((ext_vector_type(2)));
typedef float v8f __attribute__((ext_vector_type(8)));

__device__ __forceinline__ v8f wmma_f32(v2f a, v2f b, v8f c) {
  return __builtin_amdgcn_wmma_f32_16x16x4_f32(false, a, false, b, (short)0, c,
                                               false, false);
}

#if defined(__AMDGCN__) && __has_builtin(__builtin_amdgcn_global_load_async_to_lds_b128)
#define HAVE_ASYNC_LDS 1
// Builtin signature (from clang diagnostic): param1 = int vector_size(16) * in
// global (AS1) space; param2 = LDS side. Cast through uintptr_t.
typedef int v4i_vs __attribute__((vector_size(16)));
typedef __attribute__((address_space(1))) v4i_vs* as1_v4i;
typedef __attribute__((address_space(3))) v4i_vs* as3_v4i;
__device__ __forceinline__ as1_v4i to_glb(const void* p) {
  return (as1_v4i)(uintptr_t)p;
}
__device__ __forceinline__ as3_v4i to_lds(void* p) {
  return (as3_v4i)(uintptr_t)p;   // generic LDS addr: low 32 bits = LDS offset
}
__device__ __forceinline__ void wait_async0() {
#if __has_builtin(__builtin_amdgcn_s_wait_asynccnt)
  __builtin_amdgcn_s_wait_asynccnt(0);
#else
  asm volatile("s_wait_asynccnt 0" ::: "memory");
#endif
}
#endif

// ---------------------------------------------------------------------------
__global__ __launch_bounds__(256) void zero_kernel(float* __restrict__ p, int n) {
  int i = blockIdx.x * blockDim.x + threadIdx.x;
  if (i < n) p[i] = 0.0f;
}

// ---------------------------------------------------------------------------
// Kernel 1: msg = relu(x[src] + edge_attr @ We + be); aggr[dst] += msg
// One wave per 16-edge tile, 6 waves / WG (192 threads).
// sWe swizzled: row r = kk*2 + half holds {We[4kk+2h][c], We[4kk+2h+1][c]} pairs
// at r*WSTRIDE + c*2; WSTRIDE=288 => halves 32 banks apart (conflict-free b64).
// sA padded to row stride 68 => A-frag b64 reads hit 64 distinct banks.
// ---------------------------------------------------------------------------
#define EDGE_WAVES 6
#define WSTRIDE 288
#define ASTRIDE 68

__global__ __launch_bounds__(192) void gine_edge(
    const float* __restrict__ x, const float* __restrict__ ea,
    const float* __restrict__ We, const float* __restrict__ be,
    const int* __restrict__ src, const int* __restrict__ dst,
    float* __restrict__ aggr, int nTiles) {
  __shared__ __align__(16) float sWe[32 * WSTRIDE];               // 36864 B
  __shared__ __align__(16) float sA[EDGE_WAVES][16 * ASTRIDE];    // 26112 B

  const int tid  = threadIdx.x;
  const int wave = tid >> 5;
  const int lane = tid & 31;
  const int half = lane >> 4;
  const int l15  = lane & 15;

  // Stage + swizzle We: 2048 float4 over 192 threads.
  {
    const float4* w4 = reinterpret_cast<const float4*>(We);
    for (int i = tid; i < (EDIM * HID) / 4; i += 192) {
      const int k  = i >> 5;            // source row (K)
      const int c0 = (i & 31) << 2;     // source col
      const int row = ((k >> 2) << 1) + ((k >> 1) & 1);
      const int pos = k & 1;
      float4 v = w4[i];
      float* d = sWe + row * WSTRIDE + pos;
      d[(c0 + 0) * 2] = v.x;
      d[(c0 + 1) * 2] = v.y;
      d[(c0 + 2) * 2] = v.z;
      d[(c0 + 3) * 2] = v.w;
    }
  }

  const int tile  = blockIdx.x * EDGE_WAVES + wave;
  const bool valid = tile < nTiles;
  if (valid) {
    // Stage 16 edges x 64 floats into padded rows (stride 68).
    const float* gsrc = ea + (size_t)tile * 16 * EDIM;
    float* lbase = sA[wave];
#pragma unroll
    for (int j = 0; j < 8; ++j) {
      const int i   = lane + 32 * j;    // float4 index within tile
      const int row = i >> 4;
      const int c4  = i & 15;
      const float* gp = gsrc + row * EDIM + c4 * 4;
      float* lp = lbase + row * ASTRIDE + c4 * 4;
#ifdef HAVE_ASYNC_LDS
      __builtin_amdgcn_global_load_async_to_lds_b128(to_glb(gp), to_lds(lp), 0, 0);
#else
      *reinterpret_cast<float4*>(lp) = *reinterpret_cast<const float4*>(gp);
#endif
    }
  }
#ifdef HAVE_ASYNC_LDS
  wait_async0();
#endif
  __syncthreads();
  if (!valid) return;

  const float* sAw = sA[wave] + l15 * ASTRIDE + half * 2;   // A[m=l15][half*2 + ...]
  const float* sWp = sWe + half * WSTRIDE + l15 * 2;        // per-lane B base
  v8f acc[8] = {};

#pragma unroll
  for (int kk = 0; kk < EDIM / 4; ++kk) {
    v2f Af = *(const v2f*)(sAw + kk * 4);
    const float* wk = sWp + kk * 2 * WSTRIDE;
#pragma unroll
    for (int nt = 0; nt < 8; ++nt) {
      v2f Bf = *(const v2f*)(wk + nt * 32);
      acc[nt] = wmma_f32(Af, Bf, acc[nt]);
    }
  }

  float bev[8];
#pragma unroll
  for (int nt = 0; nt < 8; ++nt) bev[nt] = be[nt * 16 + l15];

  const int eBase = tile * 16 + half * 8;
#pragma unroll
  for (int v = 0; v < 8; ++v) {
    const int e = eBase + v;
    const int s = src[e];
    const int d = dst[e];
    const float* xr = x + (size_t)s * HID;
    float* ar = aggr + (size_t)d * HID;
#pragma unroll
    for (int nt = 0; nt < 8; ++nt) {
      const int col = nt * 16 + l15;
      float val = acc[nt][v] + bev[nt] + xr[col];
      val = fmaxf(val, 0.0f);
      unsafeAtomicAdd(ar + col, val);   // global_atomic_add_f32 (no return)
    }
  }
}

// ---------------------------------------------------------------------------
// Kernel 2: h0 = (1+eps)x + aggr ; h2 = relu(h0@W1+b1)@W2+b2 ; BN partial sums.
// 8 waves / WG, wave w owns output columns [16w,16w+16). Weights in registers
// as v2f pairs. sH/sG row stride 132 => conflict-free A-frag b64 reads.
// ---------------------------------------------------------------------------
#define HSTRIDE 132

__global__ __launch_bounds__(256) void gine_mlp(
    const float* __restrict__ x, const float* __restrict__ aggr,
    const float* __restrict__ W1, const float* __restrict__ b1,
    const float* __restrict__ W2, const float* __restrict__ b2,
    const float* __restrict__ epsp, float* __restrict__ h2,
    float* __restrict__ bsum, float* __restrict__ bsq, int nTiles) {
  __shared__ __align__(16) float sH[16 * HSTRIDE];   // 8448 B
  __shared__ __align__(16) float sG[16 * HSTRIDE];   // 8448 B

  const int tid  = threadIdx.x;
  const int wave = tid >> 5;
  const int lane = tid & 31;
  const int half = lane >> 4;
  const int l15  = lane & 15;
  const int colW = wave * 16 + l15;

  // Register-resident B fragments for this lane's column (even-aligned pairs).
  v2f w1f[32], w2f[32];
#pragma unroll
  for (int kk = 0; kk < 32; ++kk) {
    const int kB = kk * 4 + half * 2;
    v2f a, b;
    a.x = W1[kB * HID + colW];
    a.y = W1[(kB + 1) * HID + colW];
    b.x = W2[kB * HID + colW];
    b.y = W2[(kB + 1) * HID + colW];
    w1f[kk] = a;
    w2f[kk] = b;
  }
  const float b1c  = b1[colW];
  const float b2c  = b2[colW];
  const float eps1 = 1.0f + epsp[0];

  for (int tile = blockIdx.x; tile < nTiles; tile += gridDim.x) {
    // Stage h0 = (1+eps)*x + aggr into padded rows: 512 float4 over 256 threads.
    {
      const float4* x4 =
          reinterpret_cast<const float4*>(x + (size_t)tile * 16 * HID);
      const float4* a4 =
          reinterpret_cast<const float4*>(aggr + (size_t)tile * 16 * HID);
#pragma unroll
      for (int j = 0; j < 2; ++j) {
        const int i   = tid + 256 * j;
        const int row = i >> 5;
        const int c   = i & 31;
        float4 xv = x4[i], av = a4[i];
        float4 hv;
        hv.x = eps1 * xv.x + av.x;
        hv.y = eps1 * xv.y + av.y;
        hv.z = eps1 * xv.z + av.z;
        hv.w = eps1 * xv.w + av.w;
        *reinterpret_cast<float4*>(sH + row * HSTRIDE + c * 4) = hv;
      }
    }
    __syncthreads();

    // GEMM1: g = relu(h0 @ W1 + b1), this wave's 16-col strip.
    const float* sHp = sH + l15 * HSTRIDE + half * 2;
    v8f acc = {};
#pragma unroll
    for (int kk = 0; kk < 32; ++kk) {
      v2f Af = *(const v2f*)(sHp + kk * 4);
      acc = wmma_f32(Af, w1f[kk], acc);
    }
#pragma unroll
    for (int v = 0; v < 8; ++v) {
      float g = fmaxf(acc[v] + b1c, 0.0f);
      sG[(v + half * 8) * HSTRIDE + colW] = g;
    }
    __syncthreads();

    // GEMM2: h2 = g @ W2 + b2, plus BN partial sums.
    const float* sGp = sG + l15 * HSTRIDE + half * 2;
    v8f acc2 = {};
#pragma unroll
    for (int kk = 0; kk < 32; ++kk) {
      v2f Af = *(const v2f*)(sGp + kk * 4);
      acc2 = wmma_f32(Af, w2f[kk], acc2);
    }
    const int row0 = tile * 16 + half * 8;
    float ps = 0.0f, pq = 0.0f;
#pragma unroll
    for (int v = 0; v < 8; ++v) {
      float hv = acc2[v] + b2c;
      h2[(size_t)(row0 + v) * HID + colW] = hv;
      ps += hv;
      pq += hv * hv;
    }
    unsafeAtomicAdd(bsum + colW, ps);
    unsafeAtomicAdd(bsq + colW, pq);
  }
}

// ---------------------------------------------------------------------------
// Kernel 3: out = relu((h2-mu)*rsqrt(var+eps)*gamma + beta) + x
// ---------------------------------------------------------------------------
__global__ __launch_bounds__(256) void bn_finalize(
    const float* __restrict__ h2, const float* __restrict__ x,
    const float* __restrict__ bsum, const float* __restrict__ bsq,
    const float* __restrict__ gamma, const float* __restrict__ beta,
    float* __restrict__ out, int total, float invN) {
  int i = blockIdx.x * blockDim.x + threadIdx.x;
  if (i >= total) return;
  const int col = i & (HID - 1);
  const float mu  = bsum[col] * invN;
  const float var = bsq[col] * invN - mu * mu;
  const float sc  = gamma[col] * rsqrtf(var + BN_EPS);
  const float val = (h2[i] - mu) * sc + beta[col];
  out[i] = fmaxf(val, 0.0f) + x[i];
}

// ---------------------------------------------------------------------------
extern "C" void kernel_launch(void* const* d_in, const int* in_sizes, int n_in,
                              void* d_out, int out_size, void* d_ws,
                              size_t ws_size, hipStream_t stream) {
  (void)n_in; (void)out_size; (void)ws_size;
  const float* x    = (const float*)d_in[0];
  const float* ea   = (const float*)d_in[1];
  const float* We   = (const float*)d_in[2];
  const float* be   = (const float*)d_in[3];
  const float* W1   = (const float*)d_in[4];
  const float* b1   = (const float*)d_in[5];
  const float* W2   = (const float*)d_in[6];
  const float* b2   = (const float*)d_in[7];
  const float* gmm  = (const float*)d_in[8];
  const float* beta = (const float*)d_in[9];
  const float* epsp = (const float*)d_in[10];
  const int*   eidx = (const int*)d_in[11];

  const int N = in_sizes[0] / HID;
  const int E = in_sizes[11] / 2;
  const int* src = eidx;
  const int* dst = eidx + E;

  float* aggr = (float*)d_ws;                    // N*HID
  float* h2   = aggr + (size_t)N * HID;          // N*HID
  float* bsum = h2 + (size_t)N * HID;            // HID
  float* bsq  = bsum + HID;                      // HID

  // Re-zero accumulators every call (atomics accumulate into them).
  {
    const int n1 = N * HID;
    zero_kernel<<<(n1 + 255) / 256, 256, 0, stream>>>(aggr, n1);
    zero_kernel<<<1, 256, 0, stream>>>(bsum, 2 * HID);
  }

  const int eTiles = E / 16;   // 37500, divisible by EDGE_WAVES=6
  gine_edge<<<(eTiles + EDGE_WAVES - 1) / EDGE_WAVES, 192, 0, stream>>>(
      x, ea, We, be, src, dst, aggr, eTiles);

  const int nTiles = (N + 15) / 16;   // 3125
  int mlpWGs = nTiles < 1024 ? nTiles : 1024;
  gine_mlp<<<mlpWGs, 256, 0, stream>>>(x, aggr, W1, b1, W2, b2, epsp, h2, bsum,
                                       bsq, nTiles);

  const int total = N * HID;
  bn_finalize<<<(total + 255) / 256, 256, 0, stream>>>(
      h2, x, bsum, bsq, gmm, beta, (float*)d_out, total, 1.0f / (float)N);
}